// ProtoPNetConv_21964462752326
// MI455X (gfx1250) — compile-verified
//
#include <hip/hip_runtime.h>
#include <math.h>
#include <stdint.h>

typedef __attribute__((ext_vector_type(2))) float v2f;
typedef __attribute__((ext_vector_type(8))) float v8f;
typedef __attribute__((ext_vector_type(4))) unsigned int v4u;
typedef __attribute__((ext_vector_type(8))) int v8i;
typedef __attribute__((ext_vector_type(4))) int v4i;

namespace {
constexpr int S    = 512;
constexpr int E    = 1024;
constexpr int KP   = 4;
constexpr int P    = 96;
constexpr int KE   = KP * E;     // 4096
constexpr int NW   = 16;         // wave32s per block
constexpr int BLK  = NW * 32;    // 512 threads
constexpr int CHK  = 512;        // ke-columns per staged chunk
constexpr int NCHK = KE / CHK;   // 8 chunks
constexpr int TPW  = 32 / NW;    // h-tiles per wave (2)
}

__device__ __forceinline__ void wait_async0() {
#if __has_builtin(__builtin_amdgcn_s_wait_asynccnt)
    __builtin_amdgcn_s_wait_asynccnt(0);
#else
    asm volatile("s_wait_asynccnt 0x0" ::: "memory");
#endif
}

// TDM: load one 16-row x 512-col fp32 tile (row stride 4096) into LDS.
// D# per CDNA5 ISA; 2D -> remaining groups zero.
// This toolchain exposes the 6-arg builtin (clang-23 / therock-10.0 flavor).
__device__ __forceinline__ void tdm_load_chunk(const float* gsrc, uint32_t ldsByte) {
    const unsigned long long ga = (unsigned long long)(uintptr_t)gsrc;
    v4u g0;
    g0.x = 0x1u;                                           // count=1, user mode
    g0.y = ldsByte;                                        // lds_addr
    g0.z = (uint32_t)ga;                                   // global_addr[31:0]
    g0.w = (uint32_t)((ga >> 32) & 0x01FFFFFFu) | (2u << 30); // [56:32] | type=2
    const v8i g1 = {
        0x00020000,              // data_size=2 (4B), no mask/pad/iterate
        (int)(4096u << 16),      // tensor_dim0 = 4096 (bits[63:48])
        (int)(16u << 16),        // tensor_dim1 = 16   (bits[95:80])
        (int)(512u << 16),       // tile_dim0   = 512  (bits[127:112])
        16,                      // tile_dim1   = 16   (bits[143:128])
        4096,                    // tensor_dim0_stride (bits[191:160])
        0, 0
    };
    const v4i z4 = {0, 0, 0, 0};
    const v8i z8 = {0, 0, 0, 0, 0, 0, 0, 0};
    __builtin_amdgcn_tensor_load_to_lds(g0, g1, z4, z4, z8, 0);
}

// One workgroup = (sample n, dilation d, 16-prototype tile).
// B panel: TDM-streamed raw chunks -> LDS transpose -> double-buffered
// ke-major compute buffers (bank-conflict-free WMMA B reads).
__global__ __launch_bounds__(BLK)
void proto_dist_kernel(const float* __restrict__ emb,
                       const float* __restrict__ proto,
                       float* __restrict__ out)
{
    extern __shared__ float smem[];
    float* trans = smem;                       // [2][CHK*16] ke-major
    float* rawB  = smem + 2 * CHK * 16;        // [16][CHK] TDM landing pad
    float* pnSh  = rawB + CHK * 16;            // [16]
    float* wMin  = pnSh + 16;                  // [NW*16]
    float* outSh = wMin + NW * 16;             // [16]
    const uint32_t smemOff = (uint32_t)(uintptr_t)smem;
    const uint32_t rawOff  = smemOff + (uint32_t)(2 * CHK * 16) * 4u;

    const int wg    = blockIdx.x;              // 0..95
    const int n     = wg / 6;
    const int rem   = wg % 6;
    const int dIdx  = rem >> 1;
    const int tilep = rem & 1;
    const int d     = dIdx + 1;
    const int H     = S - d * (KP - 1);        // 509 / 506 / 503
    const int pb    = dIdx * 32 + tilep * 16;

    const int tid  = threadIdx.x;
    const int lane = tid & 31;
    const int wave = tid >> 5;
    const int col  = lane & 15;                // A-row / B-col
    const int hlf  = lane >> 4;                // 0 -> K=0,1  1 -> K=2,3
    const int c0   = hlf * 2;

    const float* protoTile = proto + (size_t)pb * KE;

    // LDS transpose: raw[j*CHK + kl] -> trans[buf][kl*16 + j] (v2f reads)
    auto transpose_chunk = [&](int buf) {
        float* dst = trans + buf * CHK * 16;
        for (int pidx = tid; pidx < (CHK / 2) * 16; pidx += BLK) {
            const int j   = pidx >> 8;                 // proto row
            const int klp = (pidx & 255) * 2;          // even local ke
            v2f v = *(const v2f*)(rawB + j * CHK + klp);
            dst[(klp + 0) * 16 + j] = v.x;
            dst[(klp + 1) * 16 + j] = v.y;
        }
    };

    if (wave == 0) tdm_load_chunk(protoTile, rawOff);    // chunk 0 -> raw
    if (tid < 16) pnSh[tid] = 0.0f;
    __syncthreads();

    // prototype squared norms from global (overlaps TDM latency)
    if (tid < 64) {
        const int j = tid >> 2, q = tid & 3;
        const float* pp = protoTile + (size_t)j * KE + q * (KE / 4);
        float s = 0.0f;
        for (int ke = 0; ke < KE / 4; ++ke) { const float v = pp[ke]; s += v * v; }
        atomicAdd(&pnSh[j], s);
    }
    if (wave == 0) __builtin_amdgcn_s_wait_tensorcnt(0);
    __syncthreads();
    transpose_chunk(0);
    __syncthreads();

    const float* embn = emb + (size_t)n * S * E;
    v8f   acc[TPW];
    float asum[TPW];
    #pragma unroll
    for (int tt = 0; tt < TPW; ++tt) {
        acc[tt]  = v8f{0.f,0.f,0.f,0.f,0.f,0.f,0.f,0.f};
        asum[tt] = 0.0f;
    }

    for (int c = 0; c < NCHK; ++c) {
        const int buf = c & 1;
        if (wave == 0 && c + 1 < NCHK)                   // DMA next raw chunk
            tdm_load_chunk(protoTile + (c + 1) * CHK, rawOff);

        const int   mk    = c >> 1;
        const int   eoff  = (c & 1) * CHK;
        const float* bufp = trans + buf * CHK * 16 + col;

        #pragma unroll
        for (int tt = 0; tt < TPW; ++tt) {
            const int tile = wave + tt * NW;             // 0..31
            const int hrow = tile * 16 + col;
            const int rs   = 4 * hrow + mk;
            const int k    = (rs >= H) + (rs >= 2 * H) + (rs >= 3 * H);
            int srow = rs - k * (H - d);
            srow = srow < 0 ? 0 : (srow > S - 1 ? S - 1 : srow);
            const float* rowp = embn + (size_t)srow * E + eoff;
            __builtin_prefetch(rowp + c0, 0, 1);         // global_prefetch_b8

            v8f   a8 = acc[tt];
            float as = asum[tt];
            #pragma unroll 4
            for (int t = 0; t < CHK / 4; ++t) {
                const int e = 4 * t + c0;
                v2f a = *(const v2f*)(rowp + e);         // (M=col, K=c0..c0+1)
                v2f b;                                   // (K=c0..c0+1, N=col)
                b.x = bufp[(4 * t + c0 + 0) * 16];
                b.y = bufp[(4 * t + c0 + 1) * 16];
                as += a.x * a.x + a.y * a.y;
                a8 = __builtin_amdgcn_wmma_f32_16x16x4_f32(
                         false, a, false, b, (short)0, a8, false, false);
            }
            acc[tt]  = a8;
            asum[tt] = as;
        }

        if (c + 1 < NCHK) {
            if (wave == 0) __builtin_amdgcn_s_wait_tensorcnt(0);
            __syncthreads();                             // raw chunk landed
            transpose_chunk(buf ^ 1);
            __syncthreads();                             // trans ready, raw free
        }
    }

    // ---- cosine normalize, mask invalid rows, min-pool ----
    const float pn = fmaxf(sqrtf(pnSh[col]), 1e-8f);
    float runMin = INFINITY;
    #pragma unroll
    for (int tt = 0; tt < TPW; ++tt) {
        const int h0 = (wave + tt * NW) * 16;
        float as = asum[tt] + __shfl_xor(asum[tt], 16, 32);
        const float xn = fmaxf(sqrtf(as), 1e-8f);
        float tmin = INFINITY;
        #pragma unroll
        for (int i = 0; i < 8; ++i) {
            const int row = i + 8 * hlf;
            const float xr = __shfl(xn, row, 32);
            float v = -acc[tt][i] / (xr * pn);
            v = ((h0 + row) < H) ? v : INFINITY;
            tmin = fminf(tmin, v);
        }
        tmin   = fminf(tmin, __shfl_xor(tmin, 16, 32));
        runMin = fminf(runMin, tmin);
    }

    if (lane < 16) wMin[wave * 16 + lane] = runMin;
    __syncthreads();
    if (tid < 16) {
        float m = INFINITY;
        for (int w = 0; w < NW; ++w) m = fminf(m, wMin[w * 16 + tid]);
        outSh[tid] = m;
    }
    __syncthreads();
    if (tid < 16) {   // async LDS -> global store of the 16 results
        const uint32_t ldsb = (uint32_t)(uintptr_t)(outSh + tid);
        const unsigned long long ga =
            (unsigned long long)(uintptr_t)(out + n * P + pb + tid);
        asm volatile("global_store_async_from_lds_b32 %0, %1, off"
                     :: "v"(ga), "v"(ldsb) : "memory");
    }
    wait_async0();
}

// class_out = prototype_distances @ fc_w.T  (16x96 @ 96x2) -- one wave32
__global__ void fc_kernel(const float* __restrict__ pd,
                          const float* __restrict__ fcw,
                          float* __restrict__ out)
{
    const int lane = threadIdx.x;
    if (lane < 32) {
        const int n = lane >> 1, c = lane & 1;
        float s = 0.0f;
        for (int p = 0; p < P; ++p)
            s += pd[n * P + p] * fcw[c * P + p];
        out[lane] = s;
    }
}

extern "C" void kernel_launch(void* const* d_in, const int* in_sizes, int n_in,
                              void* d_out, int out_size, void* d_ws, size_t ws_size,
                              hipStream_t stream) {
    const float* emb   = (const float*)d_in[0];
    // d_in[1]: mask (int64) -- unused by the reference math
    const float* proto = (const float*)d_in[2];
    const float* fcw   = (const float*)d_in[3];
    float* out = (float*)d_out;

    const size_t ldsBytes =
        (size_t)(2 * CHK * 16 + CHK * 16 + 16 + NW * 16 + 16) * sizeof(float);
    proto_dist_kernel<<<dim3(96), dim3(BLK), ldsBytes, stream>>>(emb, proto, out);
    fc_kernel<<<dim3(1), dim3(32), 0, stream>>>(out, fcw, out + 16 * P);
}